// RationalInterpolationModel_7069516169603
// MI455X (gfx1250) — compile-verified
//
#include <hip/hip_runtime.h>
#include <math.h>

#define N_POINTS 4096
#define M_EVAL   32768

typedef float v2f __attribute__((ext_vector_type(2)));
typedef float v8f __attribute__((ext_vector_type(8)));

// ---------------------------------------------------------------------------
// Kernel 1: per-node log-weight (double accumulator) + sign parity.
// log_w[j] = -sum_k log|n_j - n_k| + sum_p log((n_j - r_p)^2 + i_p^2)
// ---------------------------------------------------------------------------
__global__ void __launch_bounds__(256)
logw_kernel(const float* __restrict__ nodes,
            const float* __restrict__ pr,
            const float* __restrict__ pim,
            int npoles,
            double* __restrict__ lw,
            float* __restrict__ sgn)
{
    __shared__ float tile[256];
    const int j = blockIdx.x * 256 + threadIdx.x;
    const float xj = nodes[j];
    double acc = 0.0;
    int neg = 0;
    for (int t0 = 0; t0 < N_POINTS; t0 += 256) {
        tile[threadIdx.x] = nodes[t0 + threadIdx.x];
        __syncthreads();
#pragma unroll 8
        for (int k = 0; k < 256; ++k) {
            float diff = xj - tile[k];
            float ad = ((t0 + k) == j) ? 1.0f : fabsf(diff); // log(1)=0 on diagonal
            acc += (double)logf(ad);
            neg += (diff < 0.0f) ? 1 : 0;
        }
        __syncthreads();
    }
    double accp = 0.0;
    for (int p = 0; p < npoles; ++p) {
        float dr = xj - pr[p];
        float t  = dr * dr + pim[p] * pim[p];
        accp += (double)logf(t);        // == 2*log(sqrt(t))
    }
    lw[j]  = -acc + accp;
    sgn[j] = ((neg & 1) == 0) ? 1.0f : -1.0f;
}

// ---------------------------------------------------------------------------
// Kernel 2: w[j] = sign[j]*exp(lw[j]-lw[0])  (max-shift and /|w[0]| combined).
// Writes interleaved (vw, w) pairs so the eval kernel can fetch both with one
// ds_load_b128 per node pair.
// ---------------------------------------------------------------------------
__global__ void __launch_bounds__(256)
finalize_w(const double* __restrict__ lw, const float* __restrict__ sgn,
           const float* __restrict__ values,
           float2* __restrict__ wv)
{
    const int j = blockIdx.x * 256 + threadIdx.x;
    const float e  = expf((float)(lw[j] - lw[0]));
    const float wj = sgn[j] * e;
    float2 o;
    o.x = values[j] * wj;   // vw
    o.y = wj;               // w
    wv[j] = o;
}

// ---------------------------------------------------------------------------
// Kernel 3: barycentric evaluation via V_WMMA_F32_16X16X4_F32.
// Per wave: one tile of 16 eval points. Loop K over nodes in steps of 4.
//   A (16x4): row0 = vw slice, row1 = w slice, rows 2..15 = 0
//   B (4x16): B[k][col] = rcp(x_col - node_{n0+k})  (0 at exact hits)
//   D rows 0/1 accumulate num/den for the 16 points.
// Layouts per CDNA5 ISA 7.12.2: lane<16 holds (K0,K1), lane>=16 holds (K2,K3);
// D: acc[0]=row0(N=lane), acc[1]=row1 for lanes 0..15.
// Entire loop body is branchless (cndmask selects, mask*fma hit handling).
// ---------------------------------------------------------------------------
__global__ void __launch_bounds__(256)
eval_kernel(const float* __restrict__ x_eval,
            const float* __restrict__ nodes,
            const float2* __restrict__ wv,
            float* __restrict__ out)
{
    __shared__ __align__(16) float  s_nodes[N_POINTS];
    __shared__ __align__(16) float2 s_wv[N_POINTS];
    for (int i = threadIdx.x; i < N_POINTS; i += 256) {
        s_nodes[i] = nodes[i];
        s_wv[i]    = wv[i];
    }
    __syncthreads();

    const int lane    = threadIdx.x & 31;
    const int wave    = threadIdx.x >> 5;
    const int tileIdx = blockIdx.x * 8 + wave;
    const int col     = lane & 15;          // output column (eval point)
    const int kp      = (lane >> 4) << 1;   // K offset: 0 (lanes<16) or 2

    const int   m0 = tileIdx * 16;
    const float x  = x_eval[m0 + col];

    const bool isRow0 = (col == 0);         // this lane feeds A row 0 (vw)
    const bool isRow1 = (col == 1);         // this lane feeds A row 1 (w)

    v8f acc = {0.f, 0.f, 0.f, 0.f, 0.f, 0.f, 0.f, 0.f};
    float hnum = 0.f, hden = 0.f, hitf = 0.f;

#pragma unroll 4
    for (int n0 = 0; n0 < N_POINTS; n0 += 4) {
        const int ka = n0 + kp;             // even -> 8B/16B aligned LDS loads
        // broadcast loads: all 16 lanes of a half-wave read the same address
        float2 nn = *(const float2*)&s_nodes[ka];             // ds_load_b64
        float4 q  = *(const float4*)&s_wv[ka];                // ds_load_b128
        // q = (vw[ka], w[ka], vw[ka+1], w[ka+1])

        float d0 = x - nn.x;
        float d1 = x - nn.y;
        bool  h0 = fabsf(d0) < 1e-14f;
        bool  h1 = fabsf(d1) < 1e-14f;

        v2f b;
        b.x = h0 ? 0.0f : __builtin_amdgcn_rcpf(d0);          // v_rcp_f32
        b.y = h1 ? 0.0f : __builtin_amdgcn_rcpf(d1);

        // branchless exact-hit accumulation (mask * fma)
        float hm0 = h0 ? 1.0f : 0.0f;
        float hm1 = h1 ? 1.0f : 0.0f;
        hitf += hm0 + hm1;
        hnum  = fmaf(hm0, q.x, hnum);
        hden  = fmaf(hm0, q.y, hden);
        hnum  = fmaf(hm1, q.z, hnum);
        hden  = fmaf(hm1, q.w, hden);

        // branchless A-matrix row select
        v2f a;
        a.x = isRow0 ? q.x : (isRow1 ? q.y : 0.0f);
        a.y = isRow0 ? q.z : (isRow1 ? q.w : 0.0f);

        // 8 args: (neg_a, A, neg_b, B, c_mod, C, reuse_a, reuse_b)
        acc = __builtin_amdgcn_wmma_f32_16x16x4_f32(
                  false, a, false, b, (short)0, acc, false, false);
    }

    // Combine exact-hit accumulators across the (lane, lane+16) pair that
    // shares the same output column.
    hitf += __shfl_xor(hitf, 16, 32);
    hnum += __shfl_xor(hnum, 16, 32);
    hden += __shfl_xor(hden, 16, 32);

    if (lane < 16) {
        float num = acc[0];   // D row 0, N = lane
        float den = acc[1];   // D row 1, N = lane
        out[m0 + col] = (hitf > 0.0f) ? (hnum / hden) : (num / den);
    }
}

// ---------------------------------------------------------------------------
extern "C" void kernel_launch(void* const* d_in, const int* in_sizes, int n_in,
                              void* d_out, int out_size, void* d_ws, size_t ws_size,
                              hipStream_t stream)
{
    const float* x_eval    = (const float*)d_in[0];
    const float* values    = (const float*)d_in[1];
    const float* pole_real = (const float*)d_in[2];
    const float* pole_imag = (const float*)d_in[3];
    const float* nodes     = (const float*)d_in[4];
    float* out = (float*)d_out;
    const int npoles = in_sizes[2];

    char*   ws  = (char*)d_ws;
    double* lw  = (double*)ws;                               // 4096 * 8 B
    float*  sgn = (float*)(ws + N_POINTS * sizeof(double));  // 4096 * 4 B
    float2* wv  = (float2*)(ws + N_POINTS * sizeof(double)
                               + N_POINTS * sizeof(float));  // 4096 * 8 B

    logw_kernel<<<N_POINTS / 256, 256, 0, stream>>>(nodes, pole_real, pole_imag,
                                                    npoles, lw, sgn);
    finalize_w<<<N_POINTS / 256, 256, 0, stream>>>(lw, sgn, values, wv);
    eval_kernel<<<M_EVAL / (16 * 8), 256, 0, stream>>>(x_eval, nodes, wv, out);
}